// AxonalConnections_78889959292953
// MI455X (gfx1250) — compile-verified
//
#include <hip/hip_runtime.h>
#include <stdint.h>

typedef __attribute__((ext_vector_type(2))) float v2f;
typedef __attribute__((ext_vector_type(8))) float v8f;

#define SZ        16384   // H*W, both source and target count
#define BATCH     8
#define TT        128     // targets per block (8 waves x 16)
#define KT        32      // K elements staged per chunk
#define LSTR      36      // LDS row stride in floats (36r mod 64 distinct for r=0..15)
#define NSEG      4       // K-split factor for extra parallelism

// One 16-byte global->LDS async copy (ASYNCcnt-tracked, gfx1250).
// NT hint: adjacency is streamed exactly once.
__device__ __forceinline__ void asyncCopy16(unsigned ldsAddr, const float* gptr) {
  asm volatile("global_load_async_to_lds_b128 %0, %1, off th:TH_LOAD_NT"
               :: "v"(ldsAddr), "v"(gptr)
               : "memory");
}
__device__ __forceinline__ void waitAsync0() {
  asm volatile("s_wait_asynccnt 0x0" ::: "memory");
}

// dst: either d_out (dstSegStride==0, grid=128) or workspace partials
// (dstSegStride==BATCH*SZ, grid=128*NSEG). segLen = K elements per segment.
__global__ __launch_bounds__(256) void axon_gemm(const float* __restrict__ flat,
                                                 const float* __restrict__ adj,
                                                 float* __restrict__ dst,
                                                 int segLen, int dstSegStride) {
  __shared__ float adjS[2][TT * LSTR];   // 2*128*36*4 = 36864 B
  __shared__ float flatS[2][16 * LSTR];  // 2*16*36*4  =  4608 B (rows 8..15 stay 0)

  const int tid  = threadIdx.x;
  const int lane = tid & 31;
  const int wv   = tid >> 5;          // wave in block: 0..7
  const int mn   = lane & 15;         // M (A frag) == N (B frag) row-in-tile
  const int koff = (lane >> 4) << 1;  // lanes 16..31 hold K+2,K+3

  const int tBlk  = blockIdx.x & 127;
  const int kSeg  = blockIdx.x >> 7;
  const int tBase = tBlk * TT;
  const int kStart  = kSeg * segLen;
  const int nChunks = segLen / KT;
  float* outp = dst + (size_t)kSeg * dstSegStride;

  // Staging-thread mapping: 8 threads per row, 16B (4 floats) each -> KT=32 per row.
  const int tr  = tid >> 3;          // 0..31
  const int c4g = (tid & 7) << 2;    // float column within chunk: 0,4,..,28

  // Zero the padded batch rows (8..15) of both flat buffers exactly once.
  for (int i = tid; i < 8 * LSTR; i += 256) {
    flatS[0][8 * LSTR + i] = 0.0f;
    flatS[1][8 * LSTR + i] = 0.0f;
  }

  // Issue all async copies for one K-chunk into LDS buffer `buf`.
  auto issueChunk = [&](int kb, int buf) {
    const size_t col = (size_t)(kStart + kb) + c4g;
#pragma unroll
    for (int i = 0; i < 4; ++i) {
      const float* g = adj + (size_t)(tBase + i * 32 + tr) * SZ + col;
      const unsigned l =
          (unsigned)(uintptr_t)&adjS[buf][(i * 32 + tr) * LSTR + c4g];
      asyncCopy16(l, g);
    }
    if (tid < 64) {  // flat rows 0..7, 8 float4 per row
      const float* g = flat + (size_t)tr * SZ + col;
      const unsigned l = (unsigned)(uintptr_t)&flatS[buf][tr * LSTR + c4g];
      asyncCopy16(l, g);
    }
  };

  issueChunk(0, 0);

  v8f acc = {};
  const int aRow = mn * LSTR + koff;              // A frag base (flatS)
  const int bRow = (wv * 16 + mn) * LSTR + koff;  // B frag base (adjS)

  for (int c = 0; c < nChunks; ++c) {
    const int buf = c & 1;
    waitAsync0();        // my async copies for chunk c have landed in LDS
    __syncthreads();     // everyone's copies landed; prior reads of buf^1 done
    if (c + 1 < nChunks) issueChunk((c + 1) * KT, buf ^ 1);  // overlap w/ compute

#pragma unroll
    for (int cc = 0; cc < KT; cc += 4) {
      v2f a = *(const v2f*)&flatS[buf][aRow + cc];
      v2f b = *(const v2f*)&adjS[buf][bRow + cc];
      // D(16x16 f32) += A(16x4 f32) x B(4x16 f32)
      acc = __builtin_amdgcn_wmma_f32_16x16x4_f32(
          /*neg_a=*/false, a, /*neg_b=*/false, b,
          /*c_mod=*/(short)0, acc, /*reuse_a=*/false, /*reuse_b=*/false);
    }
  }

  // C/D layout: VGPR r, lanes 0-15 -> M=r (batch r), N=lane (target).
  // Lanes 16-31 / implicit M=r+8 are the zero-padded batch rows: skip.
  if (lane < 16) {
    const int t0 = tBase + wv * 16 + lane;
#pragma unroll
    for (int r = 0; r < BATCH; ++r)
      outp[r * SZ + t0] = acc[r];
  }
}

__global__ __launch_bounds__(256) void axon_reduce(const float* __restrict__ ws,
                                                   float* __restrict__ out) {
  const int i = blockIdx.x * 256 + threadIdx.x;
  if (i < BATCH * SZ) {
    float s = 0.0f;
#pragma unroll
    for (int k = 0; k < NSEG; ++k) s += ws[(size_t)k * (BATCH * SZ) + i];
    out[i] = s;
  }
}

extern "C" void kernel_launch(void* const* d_in, const int* in_sizes, int n_in,
                              void* d_out, int out_size, void* d_ws, size_t ws_size,
                              hipStream_t stream) {
  const float* flat = (const float*)d_in[0];  // spikes_A: 8*16384 f32
  const float* adj  = (const float*)d_in[1];  // adjacency: 16384*16384 f32
  float* out = (float*)d_out;                 // 8*16384 f32

  const size_t OUTN = (size_t)BATCH * SZ;
  // Deterministic per-environment choice: K-split x4 via workspace if it fits.
  if (ws_size >= NSEG * OUTN * sizeof(float)) {
    float* ws = (float*)d_ws;
    axon_gemm<<<dim3(128 * NSEG), dim3(256), 0, stream>>>(flat, adj, ws,
                                                          SZ / NSEG, (int)OUTN);
    axon_reduce<<<dim3((int)((OUTN + 255) / 256)), dim3(256), 0, stream>>>(ws, out);
  } else {
    axon_gemm<<<dim3(128), dim3(256), 0, stream>>>(flat, adj, out, SZ, 0);
  }
}